// BlankEmbedding_27341761806383
// MI455X (gfx1250) — compile-verified
//
#include <hip/hip_runtime.h>

// BlankEmbedding on MI455X (gfx1250).
//
// out[b,s,:] = sum_{k=0..8} w[b,s,k] * emb_table[x[b,s-k], :]
// with banded weights w derived from the blank-run pattern (scalar DP on p).
//
// Write-bandwidth-bound: 128 MB of f32 output (~5.5 us at 23.3 TB/s); the hot
// 16 MB slice of emb_table lives in the 192 MB L2, so the 1.5x gather read
// amplification of the WMMA formulation is free. The 16x24 banded combine is
// done with chained V_WMMA_F32_16X16X4_F32 (6 K-chunks of 4), A-fragments
// hoisted and reused across all 128 N-tiles of D=2048.

typedef __attribute__((ext_vector_type(2))) float v2f;
typedef __attribute__((ext_vector_type(8))) float v8f;

#define S_LEN   4096
#define D_DIM   2048
#define TILE_M  16
#define WIN     24   // 16 output rows + 8 history rows
#define KD      9    // weight taps k = 0..8
#define NBLANK  8

__launch_bounds__(256, 1)
__global__ void blank_emb_wmma(const int* __restrict__ x,
                               const float* __restrict__ emb,
                               float* __restrict__ out)
{
    __shared__ float cbuf[2][WIN][KD + 1];   // DP coefficient double buffer
    __shared__ float pbuf[32];               // p[s] for s in [i0-15, i0+14]
    __shared__ float Wt[TILE_M][WIN + 2];    // banded 16x24 weight matrix (padded)
    __shared__ int   toks[WIN];              // tokens for window rows

    const int tid   = threadIdx.x;
    const int bIdx  = blockIdx.x >> 8;          // batch (S/16 = 256 tiles/batch)
    const int i0    = (blockIdx.x & 255) << 4;  // tile start position
    const int xbase = bIdx * S_LEN;

    // ---- p[s] = first-blank-ahead indicator; zero outside [0, S-2] ----
    if (tid < 30) {
        int s = i0 - 15 + tid;
        float p = 0.f;
        if (s >= 0 && s < S_LEN - 1) {
            unsigned cur = (unsigned)x[xbase + s];
            unsigned nxt = (unsigned)x[xbase + s + 1];
            p = (nxt < 16u && cur >= 16u) ? 1.f : 0.f;
        }
        pbuf[tid] = p;
    }
    // ---- window tokens (clamped; clamped rows always carry zero weight) ----
    if (tid < WIN) {
        int s = i0 - 8 + tid;
        toks[tid] = x[xbase + (s < 0 ? 0 : s)];
    }
    // ---- c_0[j][k] = (k==0) ----
    if (tid < WIN * KD) {
        int j = tid / KD, k = tid % KD;
        cbuf[0][j][k] = (k == 0) ? 1.f : 0.f;
    }
    __syncthreads();

    // ---- DP: c_t[j][k] = c_{t-1}[j][k] + p[j+7-t] * c_{t-1}[j-1][k-1] ----
    for (int t = 1; t <= NBLANK; ++t) {
        const int src = (t - 1) & 1, dst = t & 1;
        if (tid < WIN) {
            const int j = tid;
            const float pj = (j >= 1) ? pbuf[j + 7 - t] : 0.f;
            cbuf[dst][j][0] = cbuf[src][j][0];
#pragma unroll
            for (int k = 1; k < KD; ++k) {
                float add = (j >= 1) ? pj * cbuf[src][j - 1][k - 1] : 0.f;
                cbuf[dst][j][k] = cbuf[src][j][k] + add;
            }
        }
        __syncthreads();
    }

    // ---- banded W[m][r] = c_8[m+8][m+8-r] (r = window source row) ----
    for (int idx = tid; idx < TILE_M * WIN; idx += 256) {
        int m = idx / WIN, r = idx % WIN;
        int k = m + 8 - r;
        Wt[m][r] = (k >= 0 && k <= 8) ? cbuf[NBLANK & 1][m + 8][k] : 0.f;
    }
    __syncthreads();

    // ================= WMMA phase (uniform across each wave) =================
    const int lane = tid & 31;
    const int wave = tid >> 5;        // 8 waves / block
    const int m    = lane & 15;
    const int half = lane >> 4;

    // A fragments: 16x4 f32 layout — lanes 0-15 hold K={0,1} (VGPR0,1),
    // lanes 16-31 hold K={2,3} of each chunk.
    v2f afrag[6];
#pragma unroll
    for (int cc = 0; cc < 6; ++cc) {
        afrag[cc].x = Wt[m][cc * 4 + 2 * half + 0];
        afrag[cc].y = Wt[m][cc * 4 + 2 * half + 1];
    }
    // Embedding row bases this lane needs (mirrored B layout: row K per half).
    long rbase[12];
#pragma unroll
    for (int cc = 0; cc < 6; ++cc) {
        rbase[2 * cc + 0] = (long)toks[cc * 4 + 2 * half + 0] * D_DIM;
        rbase[2 * cc + 1] = (long)toks[cc * 4 + 2 * half + 1] * D_DIM;
    }

    const long outRow = ((long)(bIdx * S_LEN + i0)) * D_DIM;

    // 128 N-tiles of 16 columns; 16 per wave.
#pragma unroll 2
    for (int it = 0; it < 16; ++it) {
        const int n0 = (wave * 16 + it) * 16;
        v8f c = {};
#pragma unroll
        for (int cc = 0; cc < 6; ++cc) {
            v2f bf;
            bf.x = emb[rbase[2 * cc + 0] + n0 + m];   // row K=2*half+0 of chunk
            bf.y = emb[rbase[2 * cc + 1] + n0 + m];   // row K=2*half+1 of chunk
            c = __builtin_amdgcn_wmma_f32_16x16x4_f32(
                    false, afrag[cc], false, bf, (short)0, c, false, false);
        }
        // C/D layout: VGPR v -> row v + 8*half, col = lane%16.
#pragma unroll
        for (int v = 0; v < 8; ++v) {
            const int Mrow = v + 8 * half;
            out[outRow + (long)Mrow * D_DIM + n0 + m] = c[v];
        }
    }
}

extern "C" void kernel_launch(void* const* d_in, const int* in_sizes, int n_in,
                              void* d_out, int out_size, void* d_ws, size_t ws_size,
                              hipStream_t stream) {
    (void)in_sizes; (void)n_in; (void)out_size; (void)d_ws; (void)ws_size;
    const int*   x   = (const int*)d_in[0];
    const float* emb = (const float*)d_in[1];
    float*       out = (float*)d_out;

    dim3 grid(4 * (S_LEN / TILE_M));   // 1024 blocks: (batch, 16-pos tile)
    blank_emb_wmma<<<grid, 256, 0, stream>>>(x, emb, out);
}